// TianmoucRecon_mem_23141283791473
// MI455X (gfx1250) — compile-verified
//
#include <hip/hip_runtime.h>
#include <math.h>

typedef __attribute__((ext_vector_type(2))) float v2f;
typedef __attribute__((ext_vector_type(8))) float v8f;

__device__ __forceinline__ v8f vzero8() {
    v8f v;
#pragma unroll
    for (int i = 0; i < 8; ++i) v[i] = 0.0f;
    return v;
}

__device__ __forceinline__ v8f wmma_f32_4(v2f a, v2f b, v8f c) {
    // D = A(16x4,f32) * B(4x16,f32) + C(16x16,f32)  -- V_WMMA_F32_16X16X4_F32
    return __builtin_amdgcn_wmma_f32_16x16x4_f32(false, a, false, b, (short)0, c,
                                                 false, false);
}

// ---------------------------------------------------------------------------
// Downsample conv: in (4,C,IH,IW) -> out (4,C,20,40), stride S, pad 1, kernel K
// w: (Cout, Cin, K, K), bias: (Cout)
// ---------------------------------------------------------------------------
__global__ void down_conv(const float* __restrict__ in, const float* __restrict__ w,
                          const float* __restrict__ bias, float* __restrict__ out,
                          int C, int K, int S, int IH, int IW) {
    int idx = blockIdx.x * blockDim.x + threadIdx.x;
    int total = 4 * C * 800;
    if (idx >= total) return;
    int ox = idx % 40; int t = idx / 40;
    int oy = t % 20;   t /= 20;
    int co = t % C;    int b = t / C;
    float acc = bias[co];
    const float* inb = in + (size_t)b * C * IH * IW;
    const float* wco = w + (size_t)co * C * K * K;
    for (int ky = 0; ky < K; ++ky) {
        int iy = oy * S - 1 + ky;
        if (iy < 0 || iy >= IH) continue;
        for (int kx = 0; kx < K; ++kx) {
            int ix = ox * S - 1 + kx;
            if (ix < 0 || ix >= IW) continue;
            const float* wp = wco + ky * K + kx;
            const float* ip = inb + (size_t)iy * IW + ix;
            for (int ci = 0; ci < C; ++ci)
                acc = fmaf(ip[(size_t)ci * IH * IW], wp[(size_t)ci * K * K], acc);
        }
    }
    out[idx] = acc;
}

// ---------------------------------------------------------------------------
// ConvTranspose (gather form) + residual + bias:
// out(4,C,OH,OW) = zs + convT(mo(4,C,20,40), w(Cin,Cout,K,K), stride S, pad 1,
//                             out_pad 1) + bias
// ---------------------------------------------------------------------------
__global__ void up_convt(const float* __restrict__ mo, const float* __restrict__ zs,
                         const float* __restrict__ w, const float* __restrict__ bias,
                         float* __restrict__ out, int C, int K, int S, int OH, int OW) {
    int idx = blockIdx.x * blockDim.x + threadIdx.x;
    int total = 4 * C * OH * OW;
    if (idx >= total) return;
    int ox = idx % OW; int t = idx / OW;
    int oy = t % OH;   t /= OH;
    int co = t % C;    int b = t / C;
    float acc = bias[co];
    const float* mob = mo + (size_t)b * C * 800;
    int ky0 = (oy + 1) % S;
    int kx0 = (ox + 1) % S;
    for (int ky = ky0; ky < K; ky += S) {
        int iy = (oy + 1 - ky) / S;               // exact multiple of S by construction
        if (iy < 0 || iy >= 20) continue;
        for (int kx = kx0; kx < K; kx += S) {
            int ix = (ox + 1 - kx) / S;
            if (ix < 0 || ix >= 40) continue;
            const float* wp = w + (size_t)co * K * K + ky * K + kx;   // + ci*C*K*K
            const float* ip = mob + (size_t)iy * 40 + ix;
            for (int ci = 0; ci < C; ++ci)
                acc = fmaf(ip[(size_t)ci * 800], wp[(size_t)ci * C * K * K], acc);
        }
    }
    out[idx] = zs[idx] + acc;
}

// ---------------------------------------------------------------------------
// Attention readout for one scale. 1 wave (32 threads) handles 16 rows of the
// (3200 x C) query matrix.  LN -> S = qn*key^T/sqrt(C) -> softmax -> mo=aff*key
// z  : (4,C,800)   mem : (C,128) row-major (== key^T)   aff : (3200,128)
// mo : (4,C,800)   add_z: fuse mo += z (scale 0)
// ---------------------------------------------------------------------------
template <int C>
__global__ __launch_bounds__(32) void attn_readout(
    const float* __restrict__ z, const float* __restrict__ mem,
    const float* __restrict__ lnw, const float* __restrict__ lnb,
    float* __restrict__ aff, float* __restrict__ mo, int add_z) {
    __shared__ float sQ[16 * C];
    __shared__ float sS[16 * 128];

    const int lane = threadIdx.x;
    const int row  = lane & 15;       // M index (and N-in-tile index)
    const int half = lane >> 4;       // 0: lanes 0-15, 1: lanes 16-31
    const int kb   = half * 2;        // K sub-offset for A/B fragments
    const int g    = blockIdx.x * 16 + row;
    const int bb   = g / 800;
    const int hw   = g % 800;
    const float* zrow = z + ((size_t)bb * C) * 800 + hw;

    // ---- LayerNorm over channel dim (eps = 1.0), two lanes per row ----
    const int c0 = half * (C / 2);
    float s = 0.0f, s2 = 0.0f;
    for (int c = 0; c < C / 2; ++c) {
        float x = zrow[(size_t)(c0 + c) * 800];
        s += x; s2 += x * x;
    }
    s  += __shfl_xor(s, 16, 32);
    s2 += __shfl_xor(s2, 16, 32);
    const float mean = s * (1.0f / C);
    const float var  = s2 * (1.0f / C) - mean * mean;
    const float inv  = rsqrtf(var + 1.0f);
    for (int c = 0; c < C / 2; ++c) {
        int cc = c0 + c;
        float x = zrow[(size_t)cc * 800];
        sQ[row * C + cc] = (x - mean) * inv * lnw[cc] + lnb[cc];
    }
    __syncthreads();

    // ---- GEMM1: S(16x128) = qn(16xC) @ mem(Cx128) ----
    v8f acc[8];
#pragma unroll
    for (int t = 0; t < 8; ++t) acc[t] = vzero8();
    for (int k0 = 0; k0 < C; k0 += 4) {
        v2f a;
        a.x = sQ[row * C + k0 + kb];
        a.y = sQ[row * C + k0 + kb + 1];
#pragma unroll
        for (int t = 0; t < 8; ++t) {
            const float* bp = mem + (size_t)(k0 + kb) * 128 + t * 16 + row;
            v2f b; b.x = bp[0]; b.y = bp[128];
            acc[t] = wmma_f32_4(a, b, acc[t]);
        }
    }
    const float rs = rsqrtf((float)C);
#pragma unroll
    for (int t = 0; t < 8; ++t)
#pragma unroll
        for (int r = 0; r < 8; ++r)
            sS[(r + half * 8) * 128 + t * 16 + row] = acc[t][r] * rs;
    __syncthreads();

    // ---- softmax over 128 slots, two lanes per row ----
    {
        const int n0 = half * 64;
        float mx = -3.0e38f;
        for (int n = 0; n < 64; ++n) mx = fmaxf(mx, sS[row * 128 + n0 + n]);
        mx = fmaxf(mx, __shfl_xor(mx, 16, 32));
        float se = 0.0f;
        for (int n = 0; n < 64; ++n) se += __expf(sS[row * 128 + n0 + n] - mx);
        se += __shfl_xor(se, 16, 32);
        const float rinv = 1.0f / se;
        for (int n = 0; n < 64; ++n) {
            float e = __expf(sS[row * 128 + n0 + n] - mx) * rinv;
            sS[row * 128 + n0 + n] = e;
            aff[(size_t)g * 128 + n0 + n] = e;
        }
    }
    __syncthreads();

    // ---- GEMM2: mo(16xC) = aff(16x128) @ key(128xC), key[k][c] = mem[c*128+k]
    constexpr int NT = C / 16;
    v8f acc2[NT];
#pragma unroll
    for (int t = 0; t < NT; ++t) acc2[t] = vzero8();
    for (int k0 = 0; k0 < 128; k0 += 4) {
        v2f a;
        a.x = sS[row * 128 + k0 + kb];
        a.y = sS[row * 128 + k0 + kb + 1];
#pragma unroll
        for (int t = 0; t < NT; ++t) {
            const float* bp = mem + (size_t)(t * 16 + row) * 128 + k0 + kb;
            v2f b; b.x = bp[0]; b.y = bp[1];
            acc2[t] = wmma_f32_4(a, b, acc2[t]);
        }
    }
#pragma unroll
    for (int t = 0; t < NT; ++t)
#pragma unroll
        for (int r = 0; r < 8; ++r) {
            int m  = r + half * 8;
            int gg = blockIdx.x * 16 + m;
            int b2 = gg / 800, hw2 = gg % 800;
            int c  = t * 16 + row;
            size_t idx = ((size_t)b2 * C + c) * 800 + hw2;
            float v = acc2[t][r];
            if (add_z) v += z[idx];
            mo[idx] = v;
        }
}

// ---------------------------------------------------------------------------
// ConvIn: st16x(4,128,20,40) = 1x1 conv of concat([z_e0, new_z0, z_f0]) with
// w (128,384).  GEMM: (3200 x 384) @ (384 x 128), 16 rows per wave.
// ---------------------------------------------------------------------------
__global__ __launch_bounds__(32) void convin_gemm(
    const float* __restrict__ ze, const float* __restrict__ nz0,
    const float* __restrict__ zf, const float* __restrict__ w,
    float* __restrict__ out) {
    __shared__ float sX[16 * 384];
    const int lane = threadIdx.x;
    const int row  = lane & 15;
    const int half = lane >> 4;
    const int kb   = half * 2;

    for (int idx = lane; idx < 16 * 384; idx += 32) {
        int m = idx / 384, k = idx % 384;
        int gg = blockIdx.x * 16 + m;
        int b = gg / 800, hw = gg % 800;
        const float* src; int c;
        if (k < 128)      { src = ze;  c = k; }
        else if (k < 256) { src = nz0; c = k - 128; }
        else              { src = zf;  c = k - 256; }
        sX[m * 384 + k] = src[((size_t)b * 128 + c) * 800 + hw];
    }
    __syncthreads();

    v8f acc[8];
#pragma unroll
    for (int t = 0; t < 8; ++t) acc[t] = vzero8();
    for (int k0 = 0; k0 < 384; k0 += 4) {
        v2f a;
        a.x = sX[row * 384 + k0 + kb];
        a.y = sX[row * 384 + k0 + kb + 1];
#pragma unroll
        for (int t = 0; t < 8; ++t) {
            const float* bp = w + (size_t)(t * 16 + row) * 384 + k0 + kb; // w[co][ci]
            v2f b; b.x = bp[0]; b.y = bp[1];
            acc[t] = wmma_f32_4(a, b, acc[t]);
        }
    }
#pragma unroll
    for (int t = 0; t < 8; ++t)
#pragma unroll
        for (int r = 0; r < 8; ++r) {
            int m  = r + half * 8;
            int gg = blockIdx.x * 16 + m;
            int b2 = gg / 800, hw2 = gg % 800;
            int co = t * 16 + row;
            out[((size_t)b2 * 128 + co) * 800 + hw2] = acc[t][r];
        }
}

// ---------------------------------------------------------------------------
extern "C" void kernel_launch(void* const* d_in, const int* in_sizes, int n_in,
                              void* d_out, int out_size, void* d_ws, size_t ws_size,
                              hipStream_t stream) {
    (void)in_sizes; (void)n_in; (void)out_size; (void)ws_size;
    // inputs: z_e0, z_f0, z_s0..z_s4, then params flattened (sorted dict keys):
    // convin_w, db[0..3], dw[0..3], ln_b[0..4], ln_w[0..4], mem[0..4], ub[0..3], uw[0..3]
    const float* z_e0 = (const float*)d_in[0];
    const float* z_f0 = (const float*)d_in[1];
    const float* z_s[5];
    for (int i = 0; i < 5; ++i) z_s[i] = (const float*)d_in[2 + i];
    const float* convin_w = (const float*)d_in[7];
    const float* db[4]; const float* dw[4];
    const float* ln_b[5]; const float* ln_w[5]; const float* mem[5];
    const float* ub[4]; const float* uw[4];
    for (int i = 0; i < 4; ++i) db[i]  = (const float*)d_in[8 + i];
    for (int i = 0; i < 4; ++i) dw[i]  = (const float*)d_in[12 + i];
    for (int i = 0; i < 5; ++i) ln_b[i] = (const float*)d_in[16 + i];
    for (int i = 0; i < 5; ++i) ln_w[i] = (const float*)d_in[21 + i];
    for (int i = 0; i < 5; ++i) mem[i]  = (const float*)d_in[26 + i];
    for (int i = 0; i < 4; ++i) ub[i]  = (const float*)d_in[31 + i];
    for (int i = 0; i < 4; ++i) uw[i]  = (const float*)d_in[35 + i];

    const int CH[5] = {128, 128, 64, 32, 16};
    const int Ks[5] = {0, 3, 5, 9, 17};
    const int Ss[5] = {0, 2, 4, 8, 16};
    const int IH[5] = {20, 40, 80, 160, 320};
    const int IW[5] = {40, 80, 160, 320, 640};

    // workspace (floats)
    float* ws = (float*)d_ws;
    float* zd[5];
    zd[0] = (float*)z_s[0];          // scale 0: no downsample (read-only use)
    zd[1] = ws;                      // 409600
    zd[2] = ws + 409600;             // 204800
    zd[3] = ws + 614400;             // 102400
    zd[4] = ws + 716800;             //  51200
    float* mo[5];
    mo[0] = ws + 768000;             // 409600 (holds new_z0 = z_s0 + mo)
    mo[1] = ws + 1177600;            // 409600
    mo[2] = ws + 1587200;            // 204800
    mo[3] = ws + 1792000;            // 102400
    mo[4] = ws + 1894400;            //  51200

    // outputs
    float* out   = (float*)d_out;
    float* st16  = out;                                  //   409600
    float* nz1   = out + 409600;                         // 1638400
    float* nz2   = out + 2048000;                        // 3276800
    float* nz3   = out + 5324800;                        // 6553600
    float* nz4   = out + 11878400;                       // 13107200
    float* aff   = out + 24985600;                       // 5*409600
    float* nz_out[5] = {nullptr, nz1, nz2, nz3, nz4};

    // 1) downsample convs (scales 1-4)
    for (int i = 1; i < 5; ++i) {
        int total = 4 * CH[i] * 800;
        down_conv<<<(total + 255) / 256, 256, 0, stream>>>(
            z_s[i], dw[i - 1], db[i - 1], zd[i], CH[i], Ks[i], Ss[i], IH[i], IW[i]);
    }

    // 2) attention readout per scale (WMMA f32 16x16x4)
    attn_readout<128><<<200, 32, 0, stream>>>(zd[0], mem[0], ln_w[0], ln_b[0],
                                              aff + 0 * 409600, mo[0], 1);
    attn_readout<128><<<200, 32, 0, stream>>>(zd[1], mem[1], ln_w[1], ln_b[1],
                                              aff + 1 * 409600, mo[1], 0);
    attn_readout<64><<<200, 32, 0, stream>>>(zd[2], mem[2], ln_w[2], ln_b[2],
                                             aff + 2 * 409600, mo[2], 0);
    attn_readout<32><<<200, 32, 0, stream>>>(zd[3], mem[3], ln_w[3], ln_b[3],
                                             aff + 3 * 409600, mo[3], 0);
    attn_readout<16><<<200, 32, 0, stream>>>(zd[4], mem[4], ln_w[4], ln_b[4],
                                             aff + 4 * 409600, mo[4], 0);

    // 3) upsample conv-transpose + residual (scales 1-4) -> outputs
    for (int i = 1; i < 5; ++i) {
        int total = 4 * CH[i] * IH[i] * IW[i];
        up_convt<<<(total + 255) / 256, 256, 0, stream>>>(
            mo[i], z_s[i], uw[i - 1], ub[i - 1], nz_out[i], CH[i], Ks[i], Ss[i],
            IH[i], IW[i]);
    }

    // 4) ConvIn 1x1 fusion at coarsest scale -> st16x
    convin_gemm<<<200, 32, 0, stream>>>(z_e0, mo[0], z_f0, convin_w, st16);
}